// BiLSTM_CRF_SoftWord_24756191494730
// MI455X (gfx1250) — compile-verified
//
#include <hip/hip_runtime.h>
#include <math.h>

// Problem constants: V=21128, E=300, H=256, L=15, WE=5, B=64, T=256
#define BB   64
#define TT   256
#define EE   300
#define D2E  600
#define HH   256
#define G4H  1024
#define LL   15

#if defined(__HIP_DEVICE_COMPILE__) && !__has_builtin(__builtin_amdgcn_wmma_f32_16x16x4_f32)
#error "wmma_f32_16x16x4_f32 builtin not available on this toolchain (device pass)"
#endif

#if defined(__HIP_DEVICE_COMPILE__) && __has_builtin(__builtin_amdgcn_global_load_async_to_lds_b64)
#define HAVE_ASYNC_LDS 1
#else
#define HAVE_ASYNC_LDS 0
#endif

typedef __attribute__((ext_vector_type(2))) float v2f;
typedef __attribute__((ext_vector_type(8))) float v8f;
typedef __attribute__((ext_vector_type(2))) int v2i;

// D = A(16x4) * B(4x16) + C(16x16), fp32, wave32.
__device__ __forceinline__ v8f wmma_f32x4(v2f a, v2f b, v8f c) {
#if defined(__HIP_DEVICE_COMPILE__)
  return __builtin_amdgcn_wmma_f32_16x16x4_f32(false, a, false, b, (short)0, c,
                                               false, false);
#else
  (void)a; (void)b;
  return c;  // host pass never executes this
#endif
}

// Async copy 8 bytes global -> LDS (per-lane scatter), tracked by ASYNCcnt.
// Builtin parameter types (from clang diagnostic): int2 ptr in AS1 (global),
// int2 ptr in AS3 (LDS), imm offset, imm cpol.
__device__ __forceinline__ void async_cp_b64(const float* g, float* l) {
#if HAVE_ASYNC_LDS
  __builtin_amdgcn_global_load_async_to_lds_b64(
      (__attribute__((address_space(1))) v2i*)g,
      (__attribute__((address_space(3))) v2i*)l, 0, 0);
#else
  *(v2f*)l = *(const v2f*)g;  // synchronous fallback
#endif
}

__device__ __forceinline__ void wait_async0() {
#if HAVE_ASYNC_LDS
#if __has_builtin(__builtin_amdgcn_s_wait_asynccnt)
  __builtin_amdgcn_s_wait_asynccnt(0);
#else
  asm volatile("s_wait_asynccnt 0" ::: "memory");
#endif
#endif
}

__device__ __forceinline__ float sigmf(float x) { return 1.0f / (1.0f + expf(-x)); }

// ---------------------------------------------------------------------------
// init: zero the cross-workgroup arrive counters every launch (ws is poisoned
// once and never re-poisoned between graph replays).
__global__ void k_init(int* cnt) {
  if (threadIdx.x < 2) cnt[threadIdx.x] = 0;
}

// ---------------------------------------------------------------------------
// Input projection GEMM with fused embedding gather:
//   G[t*64+b, n] = sum_d x(b, t_eff)[d] * Wih[n, d] + bih[n] + bhh[n]
// M = T*B = 16384, N = 4H = 1024, K = 600 (25 tiles of 24).
// LDS tiles are pair-interleaved [k/2][m][2] so a WMMA fragment is one aligned
// ds_load_b64 into an even VGPR pair (no v_mov marshalling).
// Pair-row strides: As 288 ( = 32 mod 64 -> the two half-wave K-slices land in
// disjoint bank groups), Bs 160 (same property).
__global__ __launch_bounds__(256) void k_in_proj(
    const float* __restrict__ emb, const float* __restrict__ semb,
    const int* __restrict__ ids, const int* __restrict__ sw,
    const int* __restrict__ lens, const float* __restrict__ Wih,
    const float* __restrict__ bih, const float* __restrict__ bhh,
    float* __restrict__ G, int rev) {
  __shared__ float As[12 * 288];  // 12 K-pairs x 128 rows x 2 (padded)
  __shared__ float Bs[12 * 160];  // 12 K-pairs x 64 cols x 2 (padded)
  __shared__ int eoff[128];
  __shared__ int soff[128];

  const int tid = threadIdx.x;
  const int lane = tid & 31, wv = tid >> 5;
  const int lhalf = lane >> 4, lmod = lane & 15;
  const int N0 = blockIdx.x * 64;
  const int M0 = blockIdx.y * 128;

  if (tid < 128) {
    int row = M0 + tid;
    int t = row >> 6, b = row & 63;
    if (rev) {
      int ln = lens[b];
      t = (t < ln) ? (ln - 1 - t) : t;
    }
    eoff[tid] = ids[b * TT + t] * EE;
    soff[tid] = sw[b * TT + t] * EE;
  }
  __syncthreads();

  v8f acc[4];
#pragma unroll
  for (int i = 0; i < 4; ++i) acc[i] = 0.0f;

  for (int k0 = 0; k0 < D2E; k0 += 24) {
    // stage A tile 128 rows x 24 K (float4 gathers; 16B-aligned since E%4==0
    // and k0%4==0, and a float4 never straddles the emb/soft_emb boundary)
    for (int i = tid; i < 768; i += 256) {           // 128*6 float4
      int r = i / 6, c = i % 6;
      int d0 = k0 + 4 * c;
      const float* rowp =
          (d0 < EE) ? (emb + eoff[r] + d0) : (semb + soff[r] + d0 - EE);
      float4 w = *(const float4*)rowp;
      v2f lo, hi;
      lo.x = w.x; lo.y = w.y; hi.x = w.z; hi.y = w.w;
      *(v2f*)&As[(2 * c) * 288 + r * 2] = lo;
      *(v2f*)&As[(2 * c + 1) * 288 + r * 2] = hi;
    }
    // stage B tile: Bs[k/2][n][2] = Wih[(N0+n)*600 + k0+k]
    for (int i = tid; i < 384; i += 256) {           // 64*6 float4
      int n = i / 6, c = i % 6;
      float4 w = *(const float4*)(Wih + (size_t)(N0 + n) * D2E + k0 + 4 * c);
      v2f lo, hi;
      lo.x = w.x; lo.y = w.y; hi.x = w.z; hi.y = w.w;
      *(v2f*)&Bs[(2 * c) * 160 + n * 2] = lo;
      *(v2f*)&Bs[(2 * c + 1) * 160 + n * 2] = hi;
    }
    __syncthreads();

#pragma unroll
    for (int kb = 0; kb < 24; kb += 4) {
      int p = (kb >> 1) + lhalf;  // K-pair index for this half-wave
      v2f a = *(const v2f*)&As[p * 288 + (wv * 16 + lmod) * 2];
#pragma unroll
      for (int nt = 0; nt < 4; ++nt) {
        v2f bfr = *(const v2f*)&Bs[p * 160 + (nt * 16 + lmod) * 2];
        acc[nt] = wmma_f32x4(a, bfr, acc[nt]);
      }
    }
    __syncthreads();
  }

#pragma unroll
  for (int nt = 0; nt < 4; ++nt) {
    int n = N0 + nt * 16 + lmod;
    float bias = bih[n] + bhh[n];
#pragma unroll
    for (int r = 0; r < 8; ++r) {
      int row = M0 + wv * 16 + r + 8 * lhalf;
      G[(size_t)row * G4H + n] = acc[nt][r] + bias;
    }
  }
}

// ---------------------------------------------------------------------------
// Recurrent LSTM scan. Grid = 16 blocks: dir = blk>>3 (fwd/bwd), slice = blk&7
// owns 32 hidden units. 8 waves/block: wave = (hsub<<2)|mt owns a 16(B)x16(H)
// C tile for all 4 gates; c-state lives in VGPRs across all 256 steps.
// h_prev staged pair-interleaved [k/2][b][2] (64 KB) so the A fragment is one
// ds_load_b64; staging is an async global->LDS b64 transpose-scatter (ASYNCcnt)
// when the toolchain exposes it. Cross-WG step barrier = release-add +
// acquire-spin on a per-direction counter (16 blocks co-resident).
__global__ __launch_bounds__(256) void k_lstm(
    const float* __restrict__ Gf, const float* __restrict__ Gb,
    const float* __restrict__ Whf, const float* __restrict__ Whb,
    float* __restrict__ Hf, float* __restrict__ Hr, int* __restrict__ cnt) {
  __shared__ float hS[128 * 128];  // [k/2][b][2], 64 KB

  const int dir = blockIdx.x >> 3;
  const int slice = blockIdx.x & 7;
  const float* G = dir ? Gb : Gf;
  const float* Wh = dir ? Whb : Whf;
  float* H = dir ? Hr : Hf;
  int* c = cnt + dir;

  const int tid = threadIdx.x;
  const int lane = tid & 31, wv = tid >> 5;
  const int lhalf = lane >> 4, lmod = lane & 15;
  const int mt = wv & 3;                           // batch tile (rows mt*16..)
  const int hsub = wv >> 2;                        // 0/1 within slice
  const int hcol = slice * 32 + hsub * 16 + lmod;  // this lane's H column
  const int m = mt * 16 + lmod;                    // A-frag row (batch index)

  v8f cst = 0.0f;  // LSTM cell state for this lane's 8 C-tile rows

  for (int i = tid; i < 128 * 128; i += 256) hS[i] = 0.0f;  // h_0 = 0
  __syncthreads();

  for (int t = 0; t < TT; ++t) {
    const float* Gt = G + (size_t)t * BB * G4H;
    v8f acc[4];
#pragma unroll
    for (int g = 0; g < 4; ++g) {
#pragma unroll
      for (int r = 0; r < 8; ++r)
        acc[g][r] = Gt[(size_t)(mt * 16 + r + 8 * lhalf) * G4H + g * HH + hcol];
    }
    if (t + 1 < TT)  // pull next step's gate pre-activations toward L2
      __builtin_prefetch(
          G + (size_t)(t + 1) * BB * G4H + (size_t)(mt * 16) * G4H + hcol, 0, 1);

#pragma unroll 4
    for (int k = 0; k < HH; k += 4) {
      int p = (k >> 1) + lhalf;  // K-pair index
      v2f a = *(const v2f*)&hS[p * 128 + m * 2];
#pragma unroll
      for (int g = 0; g < 4; ++g) {
        const float* wrow = Wh + (size_t)(g * HH + hcol) * HH;  // L2-resident
        v2f b = *(const v2f*)&wrow[2 * p];
        acc[g] = wmma_f32x4(a, b, acc[g]);
      }
    }

    float hv[8];
#pragma unroll
    for (int r = 0; r < 8; ++r) {
      float iv = sigmf(acc[0][r]);
      float fv = sigmf(acc[1][r]);
      float gv = tanhf(acc[2][r]);
      float ov = sigmf(acc[3][r]);
      float cv = fv * cst[r] + iv * gv;
      cst[r] = cv;
      hv[r] = ov * tanhf(cv);
    }
#pragma unroll
    for (int r = 0; r < 8; ++r) {
      int brow = mt * 16 + r + 8 * lhalf;
      H[(size_t)t * BB * HH + (size_t)brow * HH + hcol] = hv[r];
    }

    // grid sync across the 8 workgroups of this direction
    __threadfence();
    __syncthreads();
    if (tid == 0) {
      __hip_atomic_fetch_add(c, 1, __ATOMIC_RELEASE, __HIP_MEMORY_SCOPE_AGENT);
      while (__hip_atomic_load(c, __ATOMIC_ACQUIRE, __HIP_MEMORY_SCOPE_AGENT) <
             8 * (t + 1)) {
      }
    }
    __syncthreads();

    if (t + 1 < TT) {
      // stage full h_t (all slices) as [k/2][b][2]: b64-granular transpose,
      // async global->LDS when available (no VGPR round trip in the scan loop)
      const float* src = H + (size_t)t * BB * HH;
      for (int i = tid; i < 8192; i += 256) {  // 8192 h-pairs
        int p = i & 127, b = i >> 7;
        async_cp_b64(src + b * HH + 2 * p, &hS[p * 128 + b * 2]);
      }
      wait_async0();
      __syncthreads();
    }
  }
}

// ---------------------------------------------------------------------------
// emissions[b,t,:] = [h_f(b,t) ; h_b(b,t)] @ Wlin^T + blin, with
// h_b(b,t) = h_r(b, rev(b,t)). L=15 -> plain VALU dot kernel, Wlin in LDS.
__global__ __launch_bounds__(256) void k_emis(
    const float* __restrict__ Hf, const float* __restrict__ Hr,
    const int* __restrict__ lens, const float* __restrict__ Wlin,
    const float* __restrict__ blin, float* __restrict__ E) {
  __shared__ float Wl[LL * 512];
  const int b = blockIdx.x;
  const int t = threadIdx.x;
  for (int i = threadIdx.x; i < LL * 512; i += 256) Wl[i] = Wlin[i];
  __syncthreads();

  int ln = lens[b];
  int rv = (t < ln) ? (ln - 1 - t) : t;
  float acc[LL];
#pragma unroll
  for (int l = 0; l < LL; ++l) acc[l] = blin[l];

  const float* hf = Hf + (size_t)t * BB * HH + (size_t)b * HH;
  const float* hr = Hr + (size_t)rv * BB * HH + (size_t)b * HH;
  for (int k = 0; k < HH; k += 4) {
    float4 f4 = *(const float4*)(hf + k);
    float4 r4 = *(const float4*)(hr + k);
#pragma unroll
    for (int l = 0; l < LL; ++l) {
      const float* wf = Wl + l * 512 + k;
      const float* wr = Wl + l * 512 + 256 + k;
      acc[l] += f4.x * wf[0] + f4.y * wf[1] + f4.z * wf[2] + f4.w * wf[3];
      acc[l] += r4.x * wr[0] + r4.y * wr[1] + r4.z * wr[2] + r4.w * wr[3];
    }
  }
#pragma unroll
  for (int l = 0; l < LL; ++l) E[((size_t)b * TT + t) * LL + l] = acc[l];
}

// ---------------------------------------------------------------------------
// CRF NLL per batch row: one wave32 per b. Lanes 0..14 hold alpha[j]; the
// forward recursion uses cross-lane shuffles (trans column cached in VGPRs).
__global__ void k_crf(const float* __restrict__ E, const int* __restrict__ tags,
                      const int* __restrict__ lens, const float* __restrict__ st,
                      const float* __restrict__ en, const float* __restrict__ tr,
                      float* __restrict__ perb) {
  const int b = blockIdx.x;
  const int lane = threadIdx.x;  // blockDim = 32
  const int ln = lens[b];
  const int jj = (lane < LL) ? lane : 0;
  const float* Eb = E + (size_t)b * TT * LL;
  const int* tg = tags + b * TT;

  // numerator: gold-path score (parallel over t, wave reduce)
  float em = 0.f, trs = 0.f;
  for (int t = lane; t < TT; t += 32)
    if (t < ln) em += Eb[t * LL + tg[t]];
  for (int t = lane + 1; t < TT; t += 32)
    if (t < ln) trs += tr[tg[t - 1] * LL + tg[t]];
  for (int o = 16; o > 0; o >>= 1) {
    em += __shfl_down(em, o, 32);
    trs += __shfl_down(trs, o, 32);
  }
  float num = st[tg[0]] + em + trs + en[tg[ln - 1]];  // valid on lane 0

  // denominator: forward algorithm
  float trc[LL];
#pragma unroll
  for (int i = 0; i < LL; ++i) trc[i] = tr[i * LL + jj];
  float alpha = st[jj] + Eb[jj];
  for (int t = 1; t < TT; ++t) {
    if (t >= ln) break;  // mask is monotone; alpha frozen past len
    float ap = alpha;
    float mx = -3.0e38f;
#pragma unroll
    for (int i = 0; i < LL; ++i) mx = fmaxf(mx, __shfl(ap, i, 32) + trc[i]);
    float s = 0.f;
#pragma unroll
    for (int i = 0; i < LL; ++i) s += expf(__shfl(ap, i, 32) + trc[i] - mx);
    float nx = mx + logf(s) + Eb[t * LL + jj];
    alpha = (lane < LL) ? nx : alpha;
  }
  float v = (lane < LL) ? (alpha + en[jj]) : -3.0e38f;
  float mx2 = v;
  for (int o = 16; o > 0; o >>= 1) mx2 = fmaxf(mx2, __shfl_down(mx2, o, 32));
  mx2 = __shfl(mx2, 0, 32);
  float ex = (lane < LL) ? expf(v - mx2) : 0.f;
  for (int o = 16; o > 0; o >>= 1) ex += __shfl_down(ex, o, 32);
  if (lane == 0) perb[b] = num - (mx2 + logf(ex));
}

// Deterministic final reduction (no float atomics).
__global__ void k_fin(const float* __restrict__ perb, float* __restrict__ out) {
  if (threadIdx.x == 0) {
    float s = 0.f;
    for (int b = 0; b < BB; ++b) s += perb[b];
    out[0] = -s;
  }
}

// ---------------------------------------------------------------------------
extern "C" void kernel_launch(void* const* d_in, const int* in_sizes, int n_in,
                              void* d_out, int out_size, void* d_ws,
                              size_t ws_size, hipStream_t stream) {
  (void)in_sizes; (void)n_in; (void)out_size; (void)ws_size;
  const int* ids = (const int*)d_in[0];
  const int* lens = (const int*)d_in[1];
  const int* sw = (const int*)d_in[2];
  const int* tags = (const int*)d_in[3];
  const float* emb = (const float*)d_in[4];
  const float* semb = (const float*)d_in[5];
  const float* Wih_f = (const float*)d_in[6];
  const float* Whh_f = (const float*)d_in[7];
  const float* bih_f = (const float*)d_in[8];
  const float* bhh_f = (const float*)d_in[9];
  const float* Wih_b = (const float*)d_in[10];
  const float* Whh_b = (const float*)d_in[11];
  const float* bih_b = (const float*)d_in[12];
  const float* bhh_b = (const float*)d_in[13];
  const float* Wlin = (const float*)d_in[14];
  const float* blin = (const float*)d_in[15];
  const float* st = (const float*)d_in[16];
  const float* en = (const float*)d_in[17];
  const float* tr = (const float*)d_in[18];
  float* out = (float*)d_out;

  float* ws = (float*)d_ws;
  size_t o = 0;
  float* Gf = ws + o; o += (size_t)TT * BB * G4H;   // 67 MB
  float* Gb = ws + o; o += (size_t)TT * BB * G4H;   // 67 MB
  float* Hf = ws + o; o += (size_t)TT * BB * HH;    // 17 MB
  float* Hr = ws + o; o += (size_t)TT * BB * HH;    // 17 MB
  float* Em = ws + o; o += (size_t)BB * TT * LL;    // ~1 MB
  float* pb = ws + o; o += 64;
  int* cnt = (int*)(ws + o);

  k_init<<<1, 64, 0, stream>>>(cnt);
  dim3 gp(16, 128);  // N tiles x M tiles
  k_in_proj<<<gp, 256, 0, stream>>>(emb, semb, ids, sw, lens, Wih_f, bih_f,
                                    bhh_f, Gf, 0);
  k_in_proj<<<gp, 256, 0, stream>>>(emb, semb, ids, sw, lens, Wih_b, bih_b,
                                    bhh_b, Gb, 1);
  k_lstm<<<16, 256, 0, stream>>>(Gf, Gb, Whh_f, Whh_b, Hf, Hr, cnt);
  k_emis<<<64, 256, 0, stream>>>(Hf, Hr, lens, Wlin, blin, Em);
  k_crf<<<64, 32, 0, stream>>>(Em, tags, lens, st, en, tr, pb);
  k_fin<<<1, 1, 0, stream>>>(pb, out);
}